// VNCLCell_33268816675364
// MI455X (gfx1250) — compile-verified
//
#include <hip/hip_runtime.h>
#include <math.h>

// ---------------------------------------------------------------------------
// VNCL LSTM cell, fully fused for MI455X (gfx1250, wave32, WMMA f32 16x16x4).
//
// B=4096, NF=2048, NT=300, R=512, H=512.  ~92.5 GFLOP, ~116 MB unique HBM
// traffic -> memory-floor ~5us @ 23.3 TB/s.  Stage-1 products and stage-2
// projections are fused through a 64KB LDS tile so the 6x 32MB intermediates
// never touch HBM; only logits (32MB) use d_ws.  fp32 WMMA preserves
// reference numerics (compute sits far below the matrix-op ceiling, so there
// is no reason to downcast to bf16).
//
// Load-slot optimization (this round): k-outer accumulation so each A-pair
// load (2x b64) feeds 8 WMMAs and each B fragment (2x b32) feeds 2 WMMAs
// (both 16-row M groups) -> ~1.25 vmem per WMMA instead of ~3.
// ---------------------------------------------------------------------------

typedef float v2f __attribute__((ext_vector_type(2)));
typedef float v8f __attribute__((ext_vector_type(8)));

#define BSZ 4096
#define NF  2048
#define NT  300
#define RH  512
#define HD  512

#define MT      32      // batch rows per workgroup (2 x 16-row M groups)
#define THREADS 256     // 8 wave32s

// One 16x16x4 fp32 WMMA step. Per ISA layout:
//   A (16x4, MxK): lane<16 -> M=lane, K={k0,k0+1}; lane>=16 -> M=lane-16, K={k0+2,k0+3}
//   B (4x16, KxN): same K split; N = lane%16
//   C/D (16x16):   VGPR e -> M = e + 8*(lane>=16), N = lane%16
__device__ __forceinline__ v8f wmma_f32_step(v2f a, v2f b, v8f c) {
    return __builtin_amdgcn_wmma_f32_16x16x4_f32(false, a, false, b,
                                                 (short)0, c, false, false);
}

// One stage-1 GEMM pass over this wave's 8 tiles (4 n-tiles x 2 m-groups).
// MULTIPLY=false: uLDS = A@W ; MULTIPLY=true: uLDS *= A@W (same-wave LDS
// RMW, in-order, no barrier needed).
template <int K, int LD, bool MULTIPLY>
__device__ __forceinline__ void stage1_pass(
    const float* __restrict__ A, const float* __restrict__ W,
    float* uLDS, int m0, int ni0, int r, int hi)
{
    v8f t8[8];
    #pragma unroll
    for (int i = 0; i < 8; ++i) t8[i] = (v8f){};

    const size_t rowA = (size_t)(m0 + r) * LD;        // M group 0
    const size_t rowB = (size_t)(m0 + 16 + r) * LD;   // M group 1

    #pragma unroll 2
    for (int k0 = 0; k0 < K; k0 += 4) {
        const int kk = k0 + 2 * hi;
        v2f a0; a0.x = A[rowA + kk]; a0.y = A[rowA + kk + 1];
        v2f a1; a1.x = A[rowB + kk]; a1.y = A[rowB + kk + 1];
        #pragma unroll
        for (int t = 0; t < 4; ++t) {
            const int n0 = (ni0 + t) * 16;
            v2f b; b.x = W[(size_t)kk * RH + n0 + r];
                   b.y = W[(size_t)(kk + 1) * RH + n0 + r];
            t8[2 * t]     = wmma_f32_step(a0, b, t8[2 * t]);
            t8[2 * t + 1] = wmma_f32_step(a1, b, t8[2 * t + 1]);
        }
    }

    #pragma unroll
    for (int t = 0; t < 4; ++t) {
        const int nn = (ni0 + t) * 16 + r;
        #pragma unroll
        for (int mi = 0; mi < 2; ++mi) {
            #pragma unroll
            for (int e = 0; e < 8; ++e) {
                const int mm = mi * 16 + 8 * hi + e;   // C/D layout
                if (MULTIPLY) uLDS[mm * RH + nn] *= t8[2 * t + mi][e];
                else          uLDS[mm * RH + nn]  = t8[2 * t + mi][e];
            }
        }
    }
}

// u = (A1@W1).*(A2@W2) into LDS, then acc += u @ Ws.
template <int K1, int LD1, int K2, int LD2>
__device__ __forceinline__ void product_and_project(
    const float* __restrict__ A1, const float* __restrict__ W1,
    const float* __restrict__ A2, const float* __restrict__ W2,
    const float* __restrict__ Ws,
    float* uLDS, v8f* acc,
    int m0, int ni0, int r, int hi)
{
    stage1_pass<K1, LD1, false>(A1, W1, uLDS, m0, ni0, r, hi);
    stage1_pass<K2, LD2, true >(A2, W2, uLDS, m0, ni0, r, hi);
    __syncthreads();

    // ---- stage 2: acc += u(LDS) @ Ws ----
    const int mloc0 = r;        // M group 0 row in LDS
    const int mloc1 = 16 + r;   // M group 1 row in LDS
    #pragma unroll 2
    for (int k0 = 0; k0 < RH; k0 += 4) {
        const int kk = k0 + 2 * hi;
        v2f a0; a0.x = uLDS[mloc0 * RH + kk]; a0.y = uLDS[mloc0 * RH + kk + 1];
        v2f a1; a1.x = uLDS[mloc1 * RH + kk]; a1.y = uLDS[mloc1 * RH + kk + 1];
        // warm upcoming L2-resident weight rows
        if (k0 + 32 < RH)
            __builtin_prefetch(&Ws[(size_t)(kk + 32) * HD + ni0 * 16 + r], 0, 1);
        #pragma unroll
        for (int t = 0; t < 4; ++t) {
            const int n0 = (ni0 + t) * 16;
            v2f b; b.x = Ws[(size_t)kk * HD + n0 + r];
                   b.y = Ws[(size_t)(kk + 1) * HD + n0 + r];
            acc[2 * t]     = wmma_f32_step(a0, b, acc[2 * t]);
            acc[2 * t + 1] = wmma_f32_step(a1, b, acc[2 * t + 1]);
        }
    }
    __syncthreads();   // uLDS is recycled by the next branch
}

__global__ __launch_bounds__(THREADS)
void vncl_gate_kernel(const float* __restrict__ v,  const float* __restrict__ s,
                      const float* __restrict__ x,  const float* __restrict__ h,
                      const float* __restrict__ Wa, const float* __restrict__ Wb,
                      const float* __restrict__ Wc, const float* __restrict__ Ua,
                      const float* __restrict__ Ub, const float* __restrict__ Uc,
                      const float* __restrict__ Ca, const float* __restrict__ Cb,
                      const float* __restrict__ Cc, const float* __restrict__ bias,
                      float* __restrict__ logits /* [4][B][HD] */)
{
    __shared__ float uLDS[MT * RH];          // 64 KB product tile

    const int g    = blockIdx.y;             // gate
    const int m0   = blockIdx.x * MT;        // batch tile origin
    const int lane = threadIdx.x & 31;
    const int wave = threadIdx.x >> 5;
    const int r    = lane & 15;
    const int hi   = lane >> 4;              // selects K pair / M+8 half

    // wave owns 4 consecutive n-tiles x both 16-row M groups (8 C tiles)
    const int ni0 = wave * 4;

    // per-gate weight bases
    const float* Wa_g = Wa + (size_t)g * RH * RH;
    const float* Wb_g = Wb + (size_t)g * NT * RH;
    const float* Wc_g = Wc + (size_t)g * RH * HD;
    const float* Ua_g = Ua + (size_t)g * RH * RH;
    const float* Ub_g = Ub + (size_t)g * NT * RH;
    const float* Uc_g = Uc + (size_t)g * RH * HD;
    const float* Ca_g = Ca + (size_t)g * NF * RH;
    const float* Cb_g = Cb + (size_t)g * NT * RH;
    const float* Cc_g = Cc + (size_t)g * RH * HD;

    v8f acc[8];                              // logits accumulators (4 ni x 2 mi)
    #pragma unroll
    for (int t = 0; t < 8; ++t) acc[t] = (v8f){};

    // branch 1: (x @ Wa) .* (s @ Wb) @ Wc
    product_and_project<RH, RH, NT, NT>(x, Wa_g, s, Wb_g, Wc_g,
                                        uLDS, acc, m0, ni0, r, hi);
    // branch 2: (v @ Ca) .* (s @ Cb) @ Cc
    product_and_project<NF, NF, NT, NT>(v, Ca_g, s, Cb_g, Cc_g,
                                        uLDS, acc, m0, ni0, r, hi);
    // branch 3: (s @ Ub) .* (h @ Ua) @ Uc
    product_and_project<NT, NT, RH, RH>(s, Ub_g, h, Ua_g, Uc_g,
                                        uLDS, acc, m0, ni0, r, hi);

    // epilogue: + bias, write logits[g]
    const float* bg = bias + (size_t)g * HD;
    float* Lg = logits + ((size_t)g * BSZ + m0) * HD;
    #pragma unroll
    for (int t = 0; t < 4; ++t) {
        const int nn = (ni0 + t) * 16 + r;
        #pragma unroll
        for (int mi = 0; mi < 2; ++mi) {
            #pragma unroll
            for (int e = 0; e < 8; ++e) {
                const int mm = mi * 16 + 8 * hi + e;
                Lg[(size_t)mm * HD + nn] = acc[2 * t + mi][e] + bg[nn];
            }
        }
    }
}

__global__ __launch_bounds__(256)
void lstm_epilogue(const float* __restrict__ logits,
                   const float* __restrict__ c,
                   float* __restrict__ out, int n)
{
    const int idx = blockIdx.x * blockDim.x + threadIdx.x;
    if (idx >= n) return;
    const size_t stride = (size_t)BSZ * HD;
    const float li = logits[idx];
    const float lf = logits[idx + stride];
    const float lo = logits[idx + 2 * stride];
    const float lg = logits[idx + 3 * stride];
    const float ig = 1.0f / (1.0f + __expf(-li));
    const float fg = 1.0f / (1.0f + __expf(-lf));
    const float og = 1.0f / (1.0f + __expf(-lo));
    const float gg = tanhf(lg);
    const float nc = fg * c[idx] + ig * gg;
    const float nh = og * tanhf(nc);
    out[idx]          = nh;   // new_h
    out[idx + stride] = nc;   // new_c
}

extern "C" void kernel_launch(void* const* d_in, const int* in_sizes, int n_in,
                              void* d_out, int out_size, void* d_ws, size_t ws_size,
                              hipStream_t stream) {
    (void)in_sizes; (void)n_in; (void)out_size; (void)ws_size;
    const float* v  = (const float*)d_in[0];
    const float* s  = (const float*)d_in[1];
    const float* x  = (const float*)d_in[2];   // decoder_input
    const float* h  = (const float*)d_in[3];   // rnn_h
    const float* c  = (const float*)d_in[4];   // rnn_c
    const float* Wa = (const float*)d_in[5];
    const float* Wb = (const float*)d_in[6];
    const float* Wc = (const float*)d_in[7];
    const float* Ua = (const float*)d_in[8];
    const float* Ub = (const float*)d_in[9];
    const float* Uc = (const float*)d_in[10];
    const float* Ca = (const float*)d_in[11];
    const float* Cb = (const float*)d_in[12];
    const float* Cc = (const float*)d_in[13];
    const float* b  = (const float*)d_in[14];

    float* logits = (float*)d_ws;              // 4*B*HD f32 = 32 MB scratch

    dim3 grid(BSZ / MT, 4);                    // 128 batch tiles x 4 gates
    vncl_gate_kernel<<<grid, dim3(THREADS), 0, stream>>>(
        v, s, x, h, Wa, Wb, Wc, Ua, Ub, Uc, Ca, Cb, Cc, b, logits);

    const int n = BSZ * HD;
    lstm_epilogue<<<(n + 255) / 256, 256, 0, stream>>>(logits, c, (float*)d_out, n);
}